// CAVLossV1_43138651521388
// MI455X (gfx1250) — compile-verified
//
#include <hip/hip_runtime.h>
#include <hip/hip_bf16.h>
#include <math.h>

typedef __attribute__((ext_vector_type(2))) float v2f;
typedef __attribute__((ext_vector_type(8))) float v8f;

#define B_N 512
#define P_N 196
#define C_N 80
#define D_N 512
#define K_N 16
#define TAU_INV (1.0f/0.07f)

__constant__ float c_class_w[7] = {2.5f, 2.0f, 1.0f, 5.0f, 4.0f, 5.0f, 5.0f};
__constant__ float c_conf[49] = {
    1,3,2,1,1,2,1,
    3,1,2,1,1,1,1,
    2,2,1,1,1,1,1,
    1,1,1,1,1,1,1,
    1,1,1,1,1,1,2,
    2,1,1,1,1,1,1,
    1,1,1,1,2,1,1};

__device__ __forceinline__ float wave_sum32(float v) {
#pragma unroll
  for (int m = 16; m >= 1; m >>= 1) v += __shfl_xor(v, m, 32);
  return v;
}

// ---------------- init: zero the accumulators ----------------
__global__ void init_kernel(float* acc) {
  if (threadIdx.x < 8) acc[threadIdx.x] = 0.0f;
}

// ---------------- row-normalize cav (n+1e-8) and text (max(n,1e-12)) -------
// One wave per row; each lane owns a contiguous 64B slice -> b128 loads.
__global__ void norm_rows_kernel(const float* __restrict__ cav,
                                 const float* __restrict__ text,
                                 float* __restrict__ cavN,
                                 float* __restrict__ textN) {
  const int row = blockIdx.x;   // 0..159
  const int lane = threadIdx.x; // 0..31
  const bool isCav = row < C_N;
  const int r = isCav ? row : row - C_N;
  const float* src = (isCav ? cav : text) + (size_t)r * D_N;
  float* dst = (isCav ? cavN : textN) + (size_t)r * D_N;
  const float4* s4 = (const float4*)src + lane * 4;
  float4* d4 = (float4*)dst + lane * 4;
  float4 v[4];
  float s = 0.0f;
#pragma unroll
  for (int q = 0; q < 4; ++q) {
    v[q] = s4[q];
    s += v[q].x * v[q].x + v[q].y * v[q].y + v[q].z * v[q].z + v[q].w * v[q].w;
  }
  s = wave_sum32(s);
  float nrm = sqrtf(s);
  float inv = isCav ? (1.0f / (nrm + 1e-8f)) : (1.0f / fmaxf(nrm, 1e-12f));
#pragma unroll
  for (int q = 0; q < 4; ++q) {
    float4 o;
    o.x = v[q].x * inv; o.y = v[q].y * inv; o.z = v[q].z * inv; o.w = v[q].w * inv;
    d4[q] = o;
  }
}

// ---------------- classification loss (one thread per sample) --------------
__global__ void cls_kernel(const float* __restrict__ logits,
                           const int* __restrict__ labels,
                           float* acc) {
  int b = blockIdx.x * blockDim.x + threadIdx.x;
  if (b >= B_N) return;
  float x[7];
#pragma unroll
  for (int i = 0; i < 7; ++i) x[i] = logits[b * 7 + i];
  float m = x[0];
  int pred = 0;
#pragma unroll
  for (int i = 1; i < 7; ++i)
    if (x[i] > m) { m = x[i]; pred = i; }
  float se = 0.0f;
#pragma unroll
  for (int i = 0; i < 7; ++i) se += expf(x[i] - m);
  float lse = m + logf(se);
  int lab = labels[b];
  float ce = lse - x[lab];
  float pt = expf(-ce);
  float focal = powf(1.0f - pt, 2.5f);
  float sw = c_class_w[lab];
  float cw = c_conf[lab * 7 + pred];
  float max_prob = expf(x[pred] - lse);
  float pen = 1.0f + (max_prob < 0.7f ? 0.5f : 0.0f);
  float term = 0.25f * focal * ce * sw * cw * pen;
  atomicAdd(&acc[0], term);
}

// ---------------- top-16 concepts per sample + vote weights ----------------
__global__ void topc_kernel(const float* __restrict__ cs,
                            const int* __restrict__ labels,
                            const float* __restrict__ wv,
                            int* __restrict__ top_cidx,
                            float* __restrict__ dwA,
                            int* __restrict__ isPosA,
                            float* acc) {
  int b = blockIdx.x * blockDim.x + threadIdx.x;
  if (b >= B_N) return;
  const float* row = cs + (size_t)b * C_N;
  int lab = labels[b];
  float s2 = 0.0f;
  for (int c = 0; c < C_N; ++c) {
    float w = wv[lab * C_N + c];
    s2 += w * w;
  }
  float invw = 1.0f / fmaxf(sqrtf(s2), 1e-12f);
  unsigned long long m0 = 0ull, m1 = 0ull;
  float npos = 0.0f;
  for (int k = 0; k < K_N; ++k) {
    float best = -INFINITY;
    int bi = 0;
    for (int c = 0; c < C_N; ++c) {
      bool ch = (c < 64) ? ((m0 >> c) & 1ull) : ((m1 >> (c - 64)) & 1ull);
      if (!ch) {
        float v = row[c];
        if (v > best) { best = v; bi = c; }
      }
    }
    if (bi < 64) m0 |= 1ull << bi; else m1 |= 1ull << (bi - 64);
    top_cidx[b * K_N + k] = bi;
    float wn = wv[lab * C_N + bi] * invw;
    int ip = (wn > 0.1f) ? 1 : 0;
    dwA[b * K_N + k] = wn;
    isPosA[b * K_N + k] = ip;
    npos += (float)ip;
  }
  atomicAdd(&acc[3], npos);
  atomicAdd(&acc[4], 16.0f - npos);
}

// ---------------- top-16 patches per (b, selected concept) -----------------
__global__ void topp_kernel(const float* __restrict__ psim,
                            const int* __restrict__ top_cidx,
                            int* __restrict__ top_pidx) {
  __shared__ int slotOf[C_N];
  __shared__ float colv[P_N * K_N];
  const int b = blockIdx.x;
  const int lane = threadIdx.x;
  for (int c = lane; c < C_N; c += 32) slotOf[c] = -1;
  __syncthreads();
  if (lane < K_N) slotOf[top_cidx[b * K_N + lane]] = lane;
  __syncthreads();
  const float* base = psim + (size_t)b * P_N * C_N;
  const float4* base4 = (const float4*)base;
  const int nvec = (P_N * C_N) / 4;  // 3920
  for (int i = lane; i < nvec; i += 32) {
    float4 v = base4[i];
    float vv[4] = {v.x, v.y, v.z, v.w};
    int e0 = i * 4;
#pragma unroll
    for (int q = 0; q < 4; ++q) {
      int e = e0 + q;
      int p = e / C_N;
      int c = e - p * C_N;
      int s = slotOf[c];
      if (s >= 0) colv[p * K_N + s] = vv[q];
    }
  }
  __syncthreads();
  if (lane < K_N) {
    for (int j = 0; j < K_N; ++j) {
      float best = -INFINITY;
      int bi = 0;
      for (int p = 0; p < P_N; ++p) {
        float v = colv[p * K_N + lane];
        if (v > best) { best = v; bi = p; }
      }
      colv[bi * K_N + lane] = -INFINITY;  // exclude for next pass
      top_pidx[(b * K_N + lane) * K_N + j] = bi;
    }
  }
}

// ---------------- alignment loss: one wave per (b,k) -----------------------
// Each lane owns contiguous d = [lane*16, lane*16+16) -> 4x global_load_b128
// per patch (coalesced 2KB per wave) on the bandwidth-dominant 256MB path.
__global__ void align_kernel(const float* __restrict__ pfeat,
                             const int* __restrict__ top_cidx,
                             const int* __restrict__ top_pidx,
                             const float* __restrict__ cavN,
                             const float* __restrict__ dwA,
                             const int* __restrict__ isPosA,
                             float* acc) {
  const int pair = blockIdx.x;  // b*16 + k
  const int lane = threadIdx.x;
  const int b = pair >> 4;
  const int c = top_cidx[pair];
  float m[16];
#pragma unroll
  for (int t = 0; t < 16; ++t) m[t] = 0.0f;
  for (int j = 0; j < K_N; ++j) {
    int p = top_pidx[pair * K_N + j];
    const float4* f4 = (const float4*)(pfeat + ((size_t)b * P_N + p) * D_N) + lane * 4;
    float x[16];
    float s = 0.0f;
#pragma unroll
    for (int q = 0; q < 4; ++q) {
      float4 v = f4[q];
      x[4 * q + 0] = v.x; x[4 * q + 1] = v.y;
      x[4 * q + 2] = v.z; x[4 * q + 3] = v.w;
      s += v.x * v.x + v.y * v.y + v.z * v.z + v.w * v.w;
    }
    s = wave_sum32(s);
    float inv = 1.0f / fmaxf(sqrtf(s), 1e-12f);
#pragma unroll
    for (int t = 0; t < 16; ++t) m[t] += inv * x[t];
  }
  const float4* g4 = (const float4*)(cavN + (size_t)c * D_N) + lane * 4;
  float dotv = 0.0f, nm = 0.0f, ng = 0.0f;
#pragma unroll
  for (int q = 0; q < 4; ++q) {
    float4 gv = g4[q];
    float gq[4] = {gv.x, gv.y, gv.z, gv.w};
#pragma unroll
    for (int t = 0; t < 4; ++t) {
      float mm = m[4 * q + t] * (1.0f / 16.0f);
      dotv += mm * gq[t];
      nm += mm * mm;
      ng += gq[t] * gq[t];
    }
  }
  dotv = wave_sum32(dotv);
  nm = wave_sum32(nm);
  ng = wave_sum32(ng);
  if (lane == 0) {
    float sim = dotv / fmaxf(sqrtf(nm) * sqrtf(ng), 1e-8f);
    if (isPosA[pair]) atomicAdd(&acc[1], dwA[pair] * (1.0f - sim));
    else              atomicAdd(&acc[2], 0.1f * (1.0f + sim));
  }
}

// ---------------- text contrastive loss via f32 WMMA -----------------------
// One wave per batch item: D = A(16x512 cav rows) * B(512x16 = text rows^T)
// using V_WMMA_F32_16X16X4_F32, 128 K-steps.
// A layout (16x4 f32): lanes 0-15 hold M=lane, K={k0,k0+1}; lanes 16-31 hold
// M=lane-16, K={k0+2,k0+3}. B layout (4x16) mirrors with N=lane%16.
__global__ void text_kernel(const float* __restrict__ cavN,
                            const float* __restrict__ textN,
                            const int* __restrict__ topk,
                            float* acc) {
  const int b = blockIdx.x;
  const int lane = threadIdx.x;
  const int r = lane & 15;                   // M for A, N for B
  const int kofs = (lane < 16) ? 0 : 2;
  const int ridx = topk[b * K_N + r];        // concept index 0..79
  const float* arow = cavN + (size_t)ridx * D_N + kofs;
  const float* brow = textN + (size_t)ridx * D_N + kofs;
  v8f cacc = {0.0f, 0.0f, 0.0f, 0.0f, 0.0f, 0.0f, 0.0f, 0.0f};
#pragma unroll 4
  for (int k0 = 0; k0 < D_N / 4; ++k0) {
    v2f a = *(const v2f*)(arow + k0 * 4);
    v2f bb = *(const v2f*)(brow + k0 * 4);
    cacc = __builtin_amdgcn_wmma_f32_16x16x4_f32(
        /*neg_a=*/false, a, /*neg_b=*/false, bb,
        /*c_mod=*/(short)0, cacc, /*reuse_a=*/false, /*reuse_b=*/false);
  }
  // cacc[v] = G[M][N] with M = v + (lane<16 ? 0 : 8), N = lane%16.
  float contrib = 0.0f;
#pragma unroll
  for (int v = 0; v < 8; ++v) {
    float x = cacc[v] * TAU_INV;
    x = fminf(fmaxf(x, -100.0f), 100.0f);
    float rm = x;
#pragma unroll
    for (int msk = 8; msk >= 1; msk >>= 1) rm = fmaxf(rm, __shfl_xor(rm, msk, 32));
    float s = expf(x - rm);
#pragma unroll
    for (int msk = 8; msk >= 1; msk >>= 1) s += __shfl_xor(s, msk, 32);
    float lse = rm + logf(s);
    int M = v + ((lane < 16) ? 0 : 8);
    if ((lane & 15) == M) contrib += (lse - x);  // diagonal lane of this row
  }
  contrib = wave_sum32(contrib);
  if (lane == 0) atomicAdd(&acc[5], contrib);
}

// ---------------- finalize -------------------------------------------------
__global__ void finalize_kernel(const float* acc, float* out) {
  float loss_cls = acc[0] / (float)B_N;
  float loss_align = acc[1] / fmaxf(acc[3], 1.0f) + acc[2] / fmaxf(acc[4], 1.0f);
  float loss_text = acc[5] / ((float)B_N * (float)K_N);
  out[0] = loss_cls + 0.3f * loss_align + 1.0f * loss_text;
}

extern "C" void kernel_launch(void* const* d_in, const int* in_sizes, int n_in,
                              void* d_out, int out_size, void* d_ws, size_t ws_size,
                              hipStream_t stream) {
  const float* logits = (const float*)d_in[0];   // (512,7)
  const float* cs     = (const float*)d_in[1];   // (512,80)
  const int*   labels = (const int*)d_in[2];     // (512,)
  const float* psim   = (const float*)d_in[3];   // (512,196,80)
  const float* pfeat  = (const float*)d_in[4];   // (512,196,512)
  const float* cav    = (const float*)d_in[5];   // (80,512)
  const float* wv     = (const float*)d_in[6];   // (7,80)
  const float* text   = (const float*)d_in[7];   // (80,512)
  const int*   topk   = (const int*)d_in[8];     // (512,16)
  float* out = (float*)d_out;

  // Workspace layout (bytes from base):
  char* base = (char*)d_ws;
  float* acc      = (float*)base;                            // 8 floats
  int*   top_cidx = (int*)(base + 32);                       // 512*16 ints
  float* dwA      = (float*)(base + 32 + 32768);             // 512*16 floats
  int*   isPosA   = (int*)(base + 32 + 65536);               // 512*16 ints
  int*   top_pidx = (int*)(base + 32 + 98304);               // 512*16*16 ints
  float* cavN     = (float*)(base + 32 + 98304 + 524288);    // 80*512 floats
  float* textN    = cavN + C_N * D_N;                        // 80*512 floats

  init_kernel<<<1, 32, 0, stream>>>(acc);
  norm_rows_kernel<<<2 * C_N, 32, 0, stream>>>(cav, text, cavN, textN);
  cls_kernel<<<(B_N + 255) / 256, 256, 0, stream>>>(logits, labels, acc);
  topc_kernel<<<(B_N + 255) / 256, 256, 0, stream>>>(cs, labels, wv, top_cidx, dwA, isPosA, acc);
  topp_kernel<<<B_N, 32, 0, stream>>>(psim, top_cidx, top_pidx);
  align_kernel<<<B_N * K_N, 32, 0, stream>>>(pfeat, top_cidx, top_pidx, cavN, dwA, isPosA, acc);
  text_kernel<<<B_N, 32, 0, stream>>>(cavN, textN, topk, acc);
  finalize_kernel<<<1, 1, 0, stream>>>(acc, out);
}